// SMPLModel_42116449304612
// MI455X (gfx1250) — compile-verified
//
#include <hip/hip_runtime.h>
#include <hip/hip_bf16.h>
#include <math.h>

// ---------------- constants (SMPL 24-joint tree) ----------------
__constant__ int c_par[24] = {-1,0,0,0,1,2,3,4,5,6,7,8,9,9,9,12,13,14,16,17,18,19,20,21};
__constant__ int c_nb_off[24] = {0,4,7,10,13,16,19,22,25,28,30,32,34,38,41,44,46,49,52,55,58,61,64,66};
__constant__ int c_nb_len[24] = {4,3,3,3,3,3,3,3,3,2,2,2,4,3,3,2,3,3,3,3,3,3,2,2};
__constant__ int c_nb_flat[68] = {
  0,1,2,3,  0,1,4,  0,2,5,  0,3,6,  1,4,7,  2,5,8,  3,6,9,  4,7,10, 5,8,11,
  6,9, 7,10, 8,11,
  12,13,14,15, 12,13,16, 12,14,17, 12,15, 13,16,18, 14,17,19,
  16,18,20, 17,19,21, 18,20,22, 19,21,23, 20,22, 21,23};
// ROW_MASK lengths: LENS[j] = 4*len(NEIGHBOR_LIST[j])+1 for j=0..22 (faithful to reference indexing)
__constant__ int c_lens[23] = {17,13,13,13,13,13,13,13,13,9,9,9,17,13,13,9,13,13,13,13,13,13,9};

#define PREG 200
#define KJ   24
#define NV   6890
#define XC   20670          // 3*NV
#define PPAD 208            // 13 * 16
#define QK   16             // effective K: masked q rows are all zero beyond row 12

typedef __attribute__((ext_vector_type(2))) float v2f;
typedef __attribute__((ext_vector_type(8))) float v8f;

// ---------------- init accumulators ----------------
__global__ void k_init(float* acc) {
  if (threadIdx.x < 32) acc[threadIdx.x] = 0.0f;
}

// ---------------- per (p,k): Rodrigues R, local G (3x4), quaternion delta ----------------
__global__ void k_local(const float* __restrict__ theta, const float* __restrict__ Jm,
                        float* __restrict__ q, float* __restrict__ GL) {
  int idx = blockIdx.x * blockDim.x + threadIdx.x;
  if (idx >= PREG * KJ) return;
  int p = idx / KJ, k = idx % KJ;
  float tx = theta[idx*3+0], ty = theta[idx*3+1], tz = theta[idx*3+2];
  float ang = sqrtf(tx*tx + ty*ty + tz*tz);
  ang = fmaxf(ang, 1e-8f);
  float ax = tx/ang, ay = ty/ang, az = tz/ang;
  float c = cosf(ang), s = sinf(ang), ic = 1.0f - c;
  float R00 = c + ic*ax*ax,      R01 = ic*ax*ay - s*az,  R02 = ic*ax*az + s*ay;
  float R10 = ic*ay*ax + s*az,   R11 = c + ic*ay*ay,     R12 = ic*ay*az - s*ax;
  float R20 = ic*az*ax - s*ay,   R21 = ic*az*ay + s*ax,  R22 = c + ic*az*az;
  const float* Jp = Jm + (size_t)p * KJ * 3;
  float t0, t1, t2;
  if (k == 0) { t0 = Jp[0]; t1 = Jp[1]; t2 = Jp[2]; }
  else {
    int par = c_par[k];
    t0 = Jp[k*3+0] - Jp[par*3+0];
    t1 = Jp[k*3+1] - Jp[par*3+1];
    t2 = Jp[k*3+2] - Jp[par*3+2];
  }
  float* g = GL + (size_t)idx * 12;
  g[0]=R00; g[1]=R01; g[2]=R02; g[3]=t0;
  g[4]=R10; g[5]=R11; g[6]=R12; g[7]=t1;
  g[8]=R20; g[9]=R21; g[10]=R22; g[11]=t2;
  float half = 0.5f * ang;
  float sh = sinf(half);
  float* qo = q + (size_t)idx * 4;
  qo[0] = cosf(half) - 1.0f;   // delta vs theta_to_q(0) == [1,0,0,0] in fp32
  qo[1] = ax * sh; qo[2] = ay * sh; qo[3] = az * sh;
}

// ---------------- per p: kinematic chain, world G34 with -R*J offset ----------------
__global__ void k_chain(const float* __restrict__ GL, const float* __restrict__ Jm,
                        float* __restrict__ GW) {
  int p = blockIdx.x * blockDim.x + threadIdx.x;
  if (p >= PREG) return;
  float G[KJ][12];
  for (int k = 0; k < KJ; ++k) {
    const float* gl = GL + ((size_t)p * KJ + k) * 12;
    if (k == 0) {
      for (int i = 0; i < 12; ++i) G[0][i] = gl[i];
    } else {
      int par = c_par[k];
      for (int r = 0; r < 3; ++r)
        for (int c = 0; c < 4; ++c) {
          float v = G[par][r*4+0]*gl[0*4+c] + G[par][r*4+1]*gl[1*4+c] + G[par][r*4+2]*gl[2*4+c];
          if (c == 3) v += G[par][r*4+3];
          G[k][r*4+c] = v;
        }
    }
  }
  const float* Jp = Jm + (size_t)p * KJ * 3;
  for (int k = 0; k < KJ; ++k) {
    float jx = Jp[k*3+0], jy = Jp[k*3+1], jz = Jp[k*3+2];
    float* o = GW + ((size_t)p * KJ + k) * 12;
    for (int r = 0; r < 3; ++r) {
      float off = G[k][r*4+0]*jx + G[k][r*4+1]*jy + G[k][r*4+2]*jz;
      o[r*4+0] = G[k][r*4+0]; o[r*4+1] = G[k][r*4+1];
      o[r*4+2] = G[k][r*4+2]; o[r*4+3] = G[k][r*4+3] - off;
    }
  }
}

// ---------------- build masked q_scan[23][PPAD][QK] ----------------
// row r of joint j:   q_ne(NEIGHBOR_LIST[j+1]) row r, then zeroed if r >= LENS[j] (ROW_MASK)
__global__ void k_qscan(const float* __restrict__ q, const float* __restrict__ beta2,
                        float* __restrict__ qs) {
  int idx = blockIdx.x * blockDim.x + threadIdx.x;
  if (idx >= 23 * PPAD) return;
  int j = idx / PPAD, p = idx % PPAD;
  float* out = qs + (size_t)idx * QK;
  if (p >= PREG) { for (int r = 0; r < QK; ++r) out[r] = 0.0f; return; }
  int k = j + 1;
  int off = c_nb_off[k], len = c_nb_len[k];
  int L = c_lens[j];
  float b = beta2[0];
  for (int r = 0; r < QK; ++r) {
    float v = 0.0f;
    if (r < 4*len) {
      int nb = c_nb_flat[off + (r >> 2)];
      v = q[((size_t)p * KJ + nb) * 4 + (r & 3)];
    } else if (r == 4*len) {
      v = b;
    }
    if (r >= L) v = 0.0f;       // ROW_MASK folded into the A operand
    out[r] = v;
  }
}

// ---------------- pose-blend GEMM via V_WMMA_F32_16X16X4_F32 (branch-free hot loop) ----------------
// Tc[p, x] = T[p, x] + sum_j relu(A[j, x/3]) * sum_{r<16} qs_masked[j,p,r] * Kp[j,r,x]
__global__ __launch_bounds__(32) void k_pose(const float* __restrict__ qs,
                                             const float* __restrict__ Kp,
                                             const float* __restrict__ A,
                                             const float* __restrict__ T,
                                             float* __restrict__ Tc) {
  int xBase = blockIdx.x * 16;
  int pBase = blockIdx.y * 16;
  int lane  = threadIdx.x;
  int half  = lane >> 4;        // 0 or 1
  int l15   = lane & 15;
  int col   = xBase + l15;
  int colc  = col < XC ? col : XC - 1;   // clamp: OOB columns compute garbage that is never stored

  // A-frag rows (M = p) per lane, K pair selected by half; B rows base-offset by half*2
  const float* qrow = qs + (size_t)(pBase + l15) * QK + half * 2;
  const float* kcol = Kp + (size_t)(half * 2) * XC + colc;
  const float* arow = A + colc / 3;

  v8f acc = {};
  for (int j = 0; j < 23; ++j) {
    v8f c = {};
#pragma unroll
    for (int ks = 0; ks < 4; ++ks) {
      v2f a = *(const v2f*)(qrow + 4 * ks);            // aligned float2: K pair (k0, k0+1)
      v2f b;
      b.x = kcol[(size_t)(4 * ks)     * XC];
      b.y = kcol[(size_t)(4 * ks + 1) * XC];
      c = __builtin_amdgcn_wmma_f32_16x16x4_f32(false, a, false, b, (short)0, c, false, false);
    }
    float f = fmaxf(arow[0], 0.0f);                    // relu(A[j, n])
#pragma unroll
    for (int i = 0; i < 8; ++i) acc[i] += f * c[i];
    qrow += (size_t)PPAD * QK;
    kcol += (size_t)17 * XC;
    arow += NV;
    if (j < 22) __builtin_prefetch(kcol, 0, 1);        // global_prefetch_b8: next joint's B stripe
  }

  bool fullP = (pBase + 16 <= PREG);
  bool fullX = (xBase + 16 <= XC);
  if (fullP & fullX) {                                  // block-uniform fast epilogue
#pragma unroll
    for (int i = 0; i < 8; ++i) {
      size_t o = (size_t)(pBase + i + half * 8) * XC + col;   // C layout: VGPR i -> M=i / M=i+8
      Tc[o] = acc[i] + T[o];
    }
  } else {
#pragma unroll
    for (int i = 0; i < 8; ++i) {
      int p = pBase + i + half * 8;
      if (p < PREG && col < XC) {
        size_t o = (size_t)p * XC + col;
        Tc[o] = acc[i] + T[o];
      }
    }
  }
}

// ---------------- LBS + residual energy; overwrites Tc (d_out+1) with tv in place ----------------
__global__ __launch_bounds__(256) void k_lbs(const float* __restrict__ GW,
                                             const float* __restrict__ Wp,
                                             const float* __restrict__ V,
                                             float* __restrict__ Tv,   // in: T_corr, out: tv
                                             float* __restrict__ acc) {
  __shared__ float g[KJ * 12];
  __shared__ float red[256];
  int tid = threadIdx.x;
  int p = blockIdx.y;
  int n = blockIdx.x * 256 + tid;
  for (int i = tid; i < KJ * 12; i += 256) g[i] = GW[(size_t)p * KJ * 12 + i];
  __syncthreads();
  float local = 0.0f;
  if (n < NV) {
    float M[12];
#pragma unroll
    for (int m = 0; m < 12; ++m) M[m] = 0.0f;
    const float* wrow = Wp + (size_t)n * KJ;
    for (int k = 0; k < KJ; ++k) {
      float w = wrow[k];
#pragma unroll
      for (int m = 0; m < 12; ++m) M[m] += w * g[k*12 + m];
    }
    size_t base = (size_t)p * XC + (size_t)n * 3;
    float tx = Tv[base+0], ty = Tv[base+1], tz = Tv[base+2];
    float t0 = M[0]*tx + M[1]*ty + M[2]*tz  + M[3];
    float t1 = M[4]*tx + M[5]*ty + M[6]*tz  + M[7];
    float t2 = M[8]*tx + M[9]*ty + M[10]*tz + M[11];
    Tv[base+0] = t0; Tv[base+1] = t1; Tv[base+2] = t2;
    float d0 = V[base+0] - t0, d1 = V[base+1] - t1, d2 = V[base+2] - t2;
    local = d0*d0 + d1*d1 + d2*d2;
  }
  red[tid] = local; __syncthreads();
  for (int s = 128; s > 0; s >>= 1) { if (tid < s) red[tid] += red[tid + s]; __syncthreads(); }
  if (tid == 0) atomicAdd(&acc[0], red[0]);
}

// ---------------- W' energy: sum (W'-Wi)^2 -> acc[1], sum |W'| -> acc[2] ----------------
__global__ __launch_bounds__(256) void k_we(const float* __restrict__ Wp,
                                            const float* __restrict__ Wi,
                                            float* __restrict__ acc) {
  __shared__ float s2[256], sa[256];
  int tid = threadIdx.x;
  int i = blockIdx.x * 256 + tid;
  float l2 = 0.0f, la = 0.0f;
  if (i < NV * KJ) {
    float w = Wp[i], d = w - Wi[i];
    l2 = d * d; la = fabsf(w);
  }
  s2[tid] = l2; sa[tid] = la; __syncthreads();
  for (int s = 128; s > 0; s >>= 1) {
    if (tid < s) { s2[tid] += s2[tid+s]; sa[tid] += sa[tid+s]; }
    __syncthreads();
  }
  if (tid == 0) { atomicAdd(&acc[1], s2[0]); atomicAdd(&acc[2], sa[0]); }
}

// ---------------- |A| energy -> acc[3] ----------------
__global__ __launch_bounds__(256) void k_ae(const float* __restrict__ A, float* __restrict__ acc) {
  __shared__ float red[256];
  int tid = threadIdx.x;
  int i = blockIdx.x * 256 + tid;
  float l = (i < 23 * NV) ? fabsf(A[i]) : 0.0f;
  red[tid] = l; __syncthreads();
  for (int s = 128; s > 0; s >>= 1) { if (tid < s) red[tid] += red[tid+s]; __syncthreads(); }
  if (tid == 0) atomicAdd(&acc[3], red[0]);
}

// ---------------- group-lasso partials: sum_{r<LENS[j],x} Kp[j,r,x]^2 -> acc[4+j] ----------------
__global__ __launch_bounds__(256) void k_kn(const float* __restrict__ Kp, float* __restrict__ acc) {
  __shared__ float red[256];
  int tid = threadIdx.x;
  int j = blockIdx.y;
  int x = blockIdx.x * 256 + tid;
  float l = 0.0f;
  if (x < XC) {
    int L = c_lens[j];
    const float* base = Kp + (size_t)j * 17 * XC + x;
    for (int r = 0; r < L; ++r) { float v = base[(size_t)r * XC]; l += v * v; }
  }
  red[tid] = l; __syncthreads();
  for (int s = 128; s > 0; s >>= 1) { if (tid < s) red[tid] += red[tid+s]; __syncthreads(); }
  if (tid == 0) atomicAdd(&acc[4 + j], red[0]);
}

// ---------------- combine energy ----------------
__global__ void k_final(const float* __restrict__ acc, const int* __restrict__ epoch,
                        float* __restrict__ out) {
  if (threadIdx.x != 0 || blockIdx.x != 0) return;
  float g = expf(-5e-5f * (float)epoch[0]);
  float kn = 0.0f;
  for (int j = 0; j < 23; ++j) kn += sqrtf(acc[4 + j]);
  out[0] = 0.01f*g*acc[0] + 0.01f*g*acc[1] + g*acc[2] + g*acc[3] + g*kn;
}

extern "C" void kernel_launch(void* const* d_in, const int* in_sizes, int n_in,
                              void* d_out, int out_size, void* d_ws, size_t ws_size,
                              hipStream_t stream) {
  (void)in_sizes; (void)n_in; (void)out_size; (void)ws_size;
  const float* V     = (const float*)d_in[0];
  const float* T     = (const float*)d_in[1];
  const float* Jm    = (const float*)d_in[2];
  const float* theta = (const float*)d_in[3];
  const float* beta2 = (const float*)d_in[4];
  const float* Wp    = (const float*)d_in[5];
  const float* Wi    = (const float*)d_in[6];
  const float* A     = (const float*)d_in[7];
  const float* Kp    = (const float*)d_in[8];
  const int*   epoch = (const int*)d_in[9];
  float* out = (float*)d_out;
  float* ws  = (float*)d_ws;

  // workspace layout (floats): ~850 KB total
  float* q   = ws;                       // 200*24*4   = 19200
  float* GL  = ws + 19200;               // 200*24*12  = 57600
  float* GW  = ws + 76800;               // 200*24*12  = 57600
  float* qs  = ws + 134400;              // 23*208*16  = 76544
  float* acc = ws + 210944;              // 32 accumulators
  float* Tc  = out + 1;                  // staged T_corr, overwritten with tv in k_lbs

  k_init <<<1, 32, 0, stream>>>(acc);
  k_local<<<(PREG*KJ + 127)/128, 128, 0, stream>>>(theta, Jm, q, GL);
  k_chain<<<(PREG + 63)/64, 64, 0, stream>>>(GL, Jm, GW);
  k_qscan<<<(23*PPAD + 127)/128, 128, 0, stream>>>(q, beta2, qs);

  dim3 gp((XC + 15)/16, PPAD/16);        // 1292 x 13 tiles, 1 wave each
  k_pose <<<gp, 32, 0, stream>>>(qs, Kp, A, T, Tc);

  dim3 gl((NV + 255)/256, PREG);
  k_lbs  <<<gl, 256, 0, stream>>>(GW, Wp, V, Tc, acc);

  k_we   <<<(NV*KJ + 255)/256, 256, 0, stream>>>(Wp, Wi, acc);
  k_ae   <<<(23*NV + 255)/256, 256, 0, stream>>>(A, acc);
  dim3 gk((XC + 255)/256, 23);
  k_kn   <<<gk, 256, 0, stream>>>(Kp, acc);
  k_final<<<1, 1, 0, stream>>>(acc, epoch, out);
}